// DarkChannelPriorLoss_72593537237683
// MI455X (gfx1250) — compile-verified
//
#include <hip/hip_runtime.h>
#include <hip/hip_bf16.h>

// DarkChannelPriorLoss for MI455X (gfx1250).
// Mathematically: reflect-pad does not change the set of values, so the
// reference reduces to  mean_b( min_{c,h,w} x[b,c,h,w] ).
// Pure HBM-bandwidth-bound streaming min-reduction (~100.7 MB read, ~4.3 us
// roofline at 23.3 TB/s). We stream through LDS with the CDNA5 async
// global->LDS data mover (ASYNCcnt-tracked) in a 4-deep software pipeline.

#define BATCHES      32
#define PER_BATCH_F4 196608   // 3*512*512/4 float4 per batch
#define SEGS         48       // blocks per batch
#define ITERS        16       // float4 per thread
#define TPB          256      // 8 waves (wave32)

typedef int v4i __attribute__((ext_vector_type(4)));

// 16-byte async copy: global -> LDS, tracked by ASYNCcnt.
__device__ __forceinline__ void async_copy16(const void* gp, void* lp) {
#if __has_builtin(__builtin_amdgcn_global_load_async_to_lds_b128)
  __builtin_amdgcn_global_load_async_to_lds_b128((v4i*)gp, (v4i*)lp,
                                                 /*offset=*/0, /*cpol=*/0);
#else
  unsigned lds_off = (unsigned)(unsigned long long)lp;  // LDS address: low 32 bits of the pointer
  asm volatile("global_load_async_to_lds_b128 %0, %1, off"
               :
               : "v"(lds_off), "v"(gp)
               : "memory");
#endif
}

// Wait until at most N async transfers outstanding. "memory" clobber is a
// compiler fence so LDS read-back is not hoisted above the wait.
#define WAIT_ASYNC(n) asm volatile("s_wait_asynccnt %0" ::"i"(n) : "memory")

__global__ __launch_bounds__(TPB) void dark_min_kernel(
    const float4* __restrict__ in, float* __restrict__ partial) {
  __shared__ float4 stage[4][TPB];  // 16 KB staging, 4-deep pipeline
  __shared__ float  red[TPB];

  const int tid   = threadIdx.x;
  const int seg   = blockIdx.x;
  const int batch = blockIdx.y;

  const float4* src =
      in + (size_t)batch * PER_BATCH_F4 + (size_t)seg * (TPB * ITERS);

  // Prologue: put 3 tiles in flight (each wave's lanes target their own slots,
  // so only per-wave ASYNCcnt waits are needed -- no workgroup barrier).
  async_copy16(&src[0 * TPB + tid], &stage[0][tid]);
  async_copy16(&src[1 * TPB + tid], &stage[1][tid]);
  async_copy16(&src[2 * TPB + tid], &stage[2][tid]);

  float m = __builtin_inff();
#pragma unroll
  for (int i = 0; i < ITERS; ++i) {
    if (i + 3 < ITERS)
      async_copy16(&src[(i + 3) * TPB + tid], &stage[(i + 3) & 3][tid]);

    // Loads complete in order; wait so that buffer i is resident.
    if (i + 3 < ITERS)      WAIT_ASYNC(3);
    else if (i + 2 < ITERS) WAIT_ASYNC(2);
    else if (i + 1 < ITERS) WAIT_ASYNC(1);
    else                    WAIT_ASYNC(0);

    float4 v = stage[i & 3][tid];  // ds_load_b128
    m = fminf(m, fminf(fminf(v.x, v.y), fminf(v.z, v.w)));
  }

  // Block reduction (LDS tree) -> one partial min per block.
  red[tid] = m;
  __syncthreads();
#pragma unroll
  for (int s = TPB / 2; s > 0; s >>= 1) {
    if (tid < s) red[tid] = fminf(red[tid], red[tid + s]);
    __syncthreads();
  }
  if (tid == 0) partial[batch * SEGS + seg] = red[0];
}

__global__ __launch_bounds__(32) void dark_finalize(
    const float* __restrict__ partial, float* __restrict__ out) {
  __shared__ float bmin[BATCHES];
  const int t = threadIdx.x;  // one wave32, t = batch index
  float m = __builtin_inff();
#pragma unroll 4
  for (int s = 0; s < SEGS; ++s) m = fminf(m, partial[t * SEGS + s]);
  bmin[t] = m;
  __syncthreads();
  if (t == 0) {
    float acc = 0.f;
#pragma unroll
    for (int b = 0; b < BATCHES; ++b) acc += bmin[b];
    out[0] = acc * (1.0f / 32.0f);
  }
}

extern "C" void kernel_launch(void* const* d_in, const int* in_sizes, int n_in,
                              void* d_out, int out_size, void* d_ws,
                              size_t ws_size, hipStream_t stream) {
  (void)in_sizes; (void)n_in; (void)out_size; (void)ws_size;
  const float4* in      = (const float4*)d_in[0];
  float*        partial = (float*)d_ws;   // 32*48 floats = 6 KB scratch
  float*        out     = (float*)d_out;  // single float32

  dim3 grid(SEGS, BATCHES);
  dark_min_kernel<<<grid, TPB, 0, stream>>>(in, partial);
  dark_finalize<<<1, 32, 0, stream>>>(partial, out);
}